// GPU_Actor_70162585747521
// MI455X (gfx1250) — compile-verified
//
#include <hip/hip_runtime.h>
#include <hip/hip_bf16.h>
#include <math.h>

typedef float v2f __attribute__((ext_vector_type(2)));
typedef float v4f __attribute__((ext_vector_type(4)));
typedef float v8f __attribute__((ext_vector_type(8)));
typedef unsigned int v2u __attribute__((ext_vector_type(2)));

#define BATCH 16
#define NN    2048

// ---------------------------------------------------------------------------
// K0: e[b,n] = dot(expert_node[b,n,:16], W_expert), g[b,n] = dot(gpu[b,n,:8], W_gpu)
//     plus per-batch sums of e and g. One block per batch.
// ---------------------------------------------------------------------------
__global__ void eg_kernel(const float* __restrict__ expert,
                          const float* __restrict__ gpu,
                          const float* __restrict__ W_expert,
                          const float* __restrict__ W_gpu,
                          float* __restrict__ e_out,
                          float* __restrict__ g_out,
                          float* __restrict__ esum,
                          float* __restrict__ gsum) {
  const int b   = blockIdx.x;
  const int tid = threadIdx.x;
  __shared__ float red[256];

  float we[16];
#pragma unroll
  for (int d = 0; d < 16; ++d) we[d] = W_expert[d];
  float wg[8];
#pragma unroll
  for (int d = 0; d < 8; ++d) wg[d] = W_gpu[d];

  float es = 0.0f, gs = 0.0f;
  for (int n = tid; n < NN; n += 256) {
    const float* ep = expert + ((size_t)b * NN + n) * 16;
    float e = 0.0f;
#pragma unroll
    for (int d = 0; d < 16; ++d) e += ep[d] * we[d];
    e_out[(size_t)b * NN + n] = e;
    es += e;

    const float* gp = gpu + ((size_t)b * NN + n) * 8;
    float g = 0.0f;
#pragma unroll
    for (int d = 0; d < 8; ++d) g += gp[d] * wg[d];
    g_out[(size_t)b * NN + n] = g;
    gs += g;
  }

  red[tid] = es;
  __syncthreads();
  for (int s = 128; s > 0; s >>= 1) {
    if (tid < s) red[tid] += red[tid + s];
    __syncthreads();
  }
  if (tid == 0) esum[b] = red[0];
  __syncthreads();

  red[tid] = gs;
  __syncthreads();
  for (int s = 128; s > 0; s >>= 1) {
    if (tid < s) red[tid] += red[tid + s];
    __syncthreads();
  }
  if (tid == 0) gsum[b] = red[0];
}

// ---------------------------------------------------------------------------
// K1: row sums of affinity / bandwidth / traffic via V_WMMA_F32_16X16X4_F32
//     with B = ones (rowsum == GEMV with a ones vector). One wave -> 16 rows.
//     rowsum layout: [t][b][n], t in {0:aff, 1:bw, 2:traffic}
// ---------------------------------------------------------------------------
__device__ __forceinline__ float sel8(v8f c, int ci) {
  float v = c[0];
  v = (ci == 1) ? c[1] : v;
  v = (ci == 2) ? c[2] : v;
  v = (ci == 3) ? c[3] : v;
  v = (ci == 4) ? c[4] : v;
  v = (ci == 5) ? c[5] : v;
  v = (ci == 6) ? c[6] : v;
  v = (ci == 7) ? c[7] : v;
  return v;
}

__global__ void rowsum_wmma(const float* __restrict__ aff,
                            const float* __restrict__ bw,
                            const float* __restrict__ tr,
                            float* __restrict__ rowsum) {
  const int lane = threadIdx.x & 31;
  const int wave = threadIdx.x >> 5;
  const int task = blockIdx.x * (blockDim.x >> 5) + wave; // 0..6143

  const int ROWG = NN / 16; // 128 groups of 16 rows per batch
  const int t    = task / (BATCH * ROWG);
  const int rem  = task % (BATCH * ROWG);
  const int b    = rem / ROWG;
  const int n0   = (rem % ROWG) * 16;

  const float* src = (t == 0) ? aff : (t == 1) ? bw : tr;

  // A-matrix 16x4 f32 layout: lanes 0-15 hold K=0,1 ; lanes 16-31 hold K=2,3
  const int row  = n0 + (lane & 15);
  const int col0 = (lane >> 4) * 2;
  const float* p = src + ((size_t)b * NN + row) * NN + col0;

  v8f c0 = {};
  v8f c1 = {};
  v2f ones;
  ones[0] = 1.0f;
  ones[1] = 1.0f;

#pragma unroll 4
  for (int kb = 0; kb < 512; kb += 2) {   // 512 k-blocks of 4 columns
    v2f a0 = __builtin_nontemporal_load((const v2f*)(p));
    v2f a1 = __builtin_nontemporal_load((const v2f*)(p + 4));
    p += 8;
    c0 = __builtin_amdgcn_wmma_f32_16x16x4_f32(false, a0, false, ones,
                                               (short)0, c0, false, false);
    c1 = __builtin_amdgcn_wmma_f32_16x16x4_f32(false, a1, false, ones,
                                               (short)0, c1, false, false);
  }

  // C/D layout: VGPR r, lanes 0-15 -> M=r ; lanes 16-31 -> M=r+8 (any column N)
  const bool writer = (lane < 8) || (lane >= 16 && lane < 24);
  const int  ci     = (lane < 8) ? lane : (lane - 16);
  const int  m      = (lane < 8) ? lane : (lane - 8);
  if (writer) {
    float v = sel8(c0, ci) + sel8(c1, ci);
    rowsum[((size_t)t * BATCH + b) * NN + n0 + m] = v;
  }
}

// ---------------------------------------------------------------------------
// K2: hidden[b,n] = relu(W1a*expert_ctx + W1b*gpu_ctx)
// ---------------------------------------------------------------------------
__global__ void hidden_kernel(const float* __restrict__ e,
                              const float* __restrict__ g,
                              const float* __restrict__ esum,
                              const float* __restrict__ gsum,
                              const float* __restrict__ rowsum,
                              const float* __restrict__ w_eatt,
                              const float* __restrict__ w_gatt,
                              const float* __restrict__ W1,
                              float* __restrict__ hidden) {
  const int idx = blockIdx.x * blockDim.x + threadIdx.x; // 0..32767
  const int b   = idx >> 11;

  const float wa  = w_eatt[0], wb = w_eatt[1], wc = w_eatt[2];
  const float ga  = w_gatt[0], gb = w_gatt[1];
  const float gbw = w_gatt[2], gtr = w_gatt[3];
  const float W1a = W1[0], W1b = W1[1];

  const float SA = rowsum[idx];
  const float SB = rowsum[(size_t)BATCH * NN + idx];
  const float ST = rowsum[(size_t)2 * BATCH * NN + idx];

  const float ec = 2048.0f * e[idx] * wa + wb * esum[b] + wc * SA;
  const float gc = 2048.0f * g[idx] * ga + gb * gsum[b] + gbw * SB + gtr * ST;
  const float h  = fmaxf(0.0f, W1a * ec + W1b * gc);
  hidden[idx] = h;
}

// ---------------------------------------------------------------------------
// K3: out[b,n,g] = softmax_g( mask ? -inf : hidden[b,n] * W2[g] )
//     One block per (b,n) row; thread t owns g = [8t, 8t+8).
// ---------------------------------------------------------------------------
__global__ void softmax_kernel(const float* __restrict__ hidden,
                               const unsigned char* __restrict__ mask,
                               const float* __restrict__ W2,
                               float* __restrict__ out) {
  const int row = blockIdx.x;   // b*2048 + n
  const int tid = threadIdx.x;  // 256 threads
  const int g0  = tid * 8;

  __shared__ float red[256];
  __shared__ float bc[2];

  const float h = hidden[row];
  const unsigned char* mrow = mask + (size_t)row * NN;
  float* orow = out + (size_t)row * NN;

  // W2 stays L2-resident (8KB, reused by every block): regular-temporal loads.
  v4f w2a = *(const v4f*)(W2 + g0);
  v4f w2b = *(const v4f*)(W2 + g0 + 4);
  // mask row is stream-once: non-temporal 8-byte load.
  v2u mb  = __builtin_nontemporal_load((const v2u*)(mrow + g0));
  unsigned long long bits = ((unsigned long long)mb[1] << 32) | (unsigned long long)mb[0];

  float s[8];
  float mx = -INFINITY;
#pragma unroll
  for (int i = 0; i < 8; ++i) {
    float w = (i < 4) ? w2a[i & 3] : w2b[i & 3];
    bool masked = ((bits >> (8 * i)) & 0xffULL) != 0ULL;
    s[i] = masked ? -INFINITY : h * w;
    mx = fmaxf(mx, s[i]);
  }

  // block-reduce max
  red[tid] = mx;
  __syncthreads();
  for (int st = 128; st > 0; st >>= 1) {
    if (tid < st) red[tid] = fmaxf(red[tid], red[tid + st]);
    __syncthreads();
  }
  if (tid == 0) bc[0] = red[0];
  __syncthreads();
  const float M = bc[0];

  float v[8];
  float sum = 0.0f;
#pragma unroll
  for (int i = 0; i < 8; ++i) {
    v[i] = expf(s[i] - M);  // expf(-inf - M) == 0 for masked entries
    sum += v[i];
  }

  // block-reduce sum
  red[tid] = sum;
  __syncthreads();
  for (int st = 128; st > 0; st >>= 1) {
    if (tid < st) red[tid] += red[tid + st];
    __syncthreads();
  }
  if (tid == 0) bc[1] = 1.0f / red[0];
  __syncthreads();
  const float inv = bc[1];

  v4f o0, o1;
#pragma unroll
  for (int i = 0; i < 4; ++i) o0[i] = v[i] * inv;
#pragma unroll
  for (int i = 0; i < 4; ++i) o1[i] = v[4 + i] * inv;
  __builtin_nontemporal_store(o0, (v4f*)(orow + g0));
  __builtin_nontemporal_store(o1, (v4f*)(orow + g0 + 4));
}

// ---------------------------------------------------------------------------
// Launch
// ---------------------------------------------------------------------------
extern "C" void kernel_launch(void* const* d_in, const int* in_sizes, int n_in,
                              void* d_out, int out_size, void* d_ws, size_t ws_size,
                              hipStream_t stream) {
  const float* expert_node = (const float*)d_in[0];
  const float* gpu_nodes   = (const float*)d_in[1];
  const float* affinity    = (const float*)d_in[2];
  const float* bandwidth   = (const float*)d_in[3];
  const float* traffic     = (const float*)d_in[4];
  const unsigned char* mask = (const unsigned char*)d_in[5];
  const float* W_expert    = (const float*)d_in[6];
  const float* W_gpu       = (const float*)d_in[7];
  const float* w_eatt      = (const float*)d_in[8];
  const float* w_gatt      = (const float*)d_in[9];
  const float* W_actor1    = (const float*)d_in[10];
  const float* W_actor2    = (const float*)d_in[11];
  float* out = (float*)d_out;

  // workspace layout (floats)
  float* ws      = (float*)d_ws;
  float* rowsum  = ws;                         // 3 * 16 * 2048 = 98304
  float* e_buf   = ws + 98304;                 // 32768
  float* g_buf   = ws + 131072;                // 32768
  float* hid_buf = ws + 163840;                // 32768
  float* esum    = ws + 196608;                // 16
  float* gsum    = ws + 196624;                // 16

  eg_kernel<<<BATCH, 256, 0, stream>>>(expert_node, gpu_nodes, W_expert, W_gpu,
                                       e_buf, g_buf, esum, gsum);

  // 3 tensors * 16 batches * 128 row-groups = 6144 wave tasks, 8 waves/block
  rowsum_wmma<<<768, 256, 0, stream>>>(affinity, bandwidth, traffic, rowsum);

  hidden_kernel<<<128, 256, 0, stream>>>(e_buf, g_buf, esum, gsum, rowsum,
                                         w_eatt, w_gatt, W_actor1, hid_buf);

  softmax_kernel<<<BATCH * NN, 256, 0, stream>>>(hid_buf, mask, W_actor2, out);
}